// InformationPreservingNorm_45483703664818
// MI455X (gfx1250) — compile-verified
//
#include <hip/hip_runtime.h>
#include <hip/hip_bf16.h>

#define EPS 1e-5f

typedef __attribute__((ext_vector_type(2))) float v2f;
typedef __attribute__((ext_vector_type(8))) float v8f;

static constexpr int B       = 32;
static constexpr int N       = 4096;
static constexpr int D       = 254;
static constexpr int ELEMS   = N * D;         // 1,040,384 per batch sample
static constexpr int ELEMS4  = ELEMS / 4;     // 260,096 float4 = 256 * 8 * 127
static constexpr int BPB     = 127;           // blocks per batch sample
static constexpr int THREADS = 256;           // 8 waves (wave32)
static constexpr int CHUNK4  = ELEMS4 / BPB;  // 2,048 float4 per block (exact)
static constexpr int NITER   = CHUNK4 / THREADS; // 8 full iterations, no tail

// ---------------------------------------------------------------------------
// Stage 1: per-block partial sum and sum-of-squares via V_WMMA_F32_16X16X4_F32
// (B = ones => each WMMA accumulates 64 fp32 values per wave, full fp32
// precision). Partitioning is exact: hot loop has NO guards, EXEC is
// statically all-1s for every WMMA as the ISA requires.
// ---------------------------------------------------------------------------
__global__ void __launch_bounds__(THREADS)
ipn_stage1(const float* __restrict__ x,
           float* __restrict__ sums,   // [B*BPB]
           float* __restrict__ sqs)    // [B*BPB]
{
    const int b   = blockIdx.x / BPB;
    const int blk = blockIdx.x % BPB;
    const int tid = threadIdx.x;

    const float4* __restrict__ xb =
        reinterpret_cast<const float4*>(x + (size_t)b * ELEMS);

    const int start4 = blk * CHUNK4;

    v8f c_s = {};                 // sum accumulator (16x16 f32 tile)
    v8f c_q = {};                 // sum-of-squares accumulator
    v2f ones; ones[0] = 1.0f; ones[1] = 1.0f;

#pragma unroll
    for (int it = 0; it < NITER; ++it) {
        const int i4 = start4 + it * THREADS + tid;
        const float4 v = xb[i4];
        __builtin_prefetch(&xb[i4 + THREADS], 0, 1); // global_prefetch_b8 (speculative-safe)

        v2f a0; a0[0] = v.x;        a0[1] = v.y;
        v2f a1; a1[0] = v.z;        a1[1] = v.w;
        v2f q0; q0[0] = v.x * v.x;  q0[1] = v.y * v.y;
        v2f q1; q1[0] = v.z * v.z;  q1[1] = v.w * v.w;

        c_s = __builtin_amdgcn_wmma_f32_16x16x4_f32(false, a0, false, ones,
                                                    (short)0, c_s, false, false);
        c_s = __builtin_amdgcn_wmma_f32_16x16x4_f32(false, a1, false, ones,
                                                    (short)0, c_s, false, false);
        c_q = __builtin_amdgcn_wmma_f32_16x16x4_f32(false, q0, false, ones,
                                                    (short)0, c_q, false, false);
        c_q = __builtin_amdgcn_wmma_f32_16x16x4_f32(false, q1, false, ones,
                                                    (short)0, c_q, false, false);
    }

    // Every column of the D tile holds identical row-sums. Fold the 8 row
    // VGPRs (rows 0-7 on lanes 0-15, rows 8-15 on lanes 16-31), then fold
    // the two half-waves.
    float s = c_s[0] + c_s[1] + c_s[2] + c_s[3] + c_s[4] + c_s[5] + c_s[6] + c_s[7];
    float q = c_q[0] + c_q[1] + c_q[2] + c_q[3] + c_q[4] + c_q[5] + c_q[6] + c_q[7];
    s += __shfl_xor(s, 16, 32);
    q += __shfl_xor(q, 16, 32);

    __shared__ float sbuf[THREADS / 32];
    __shared__ float qbuf[THREADS / 32];
    const int wave = tid >> 5;
    const int lane = tid & 31;
    if (lane == 0) { sbuf[wave] = s; qbuf[wave] = q; }
    __syncthreads();
    if (tid == 0) {
        float ts = 0.0f, tq = 0.0f;
#pragma unroll
        for (int wv = 0; wv < THREADS / 32; ++wv) { ts += sbuf[wv]; tq += qbuf[wv]; }
        sums[b * BPB + blk] = ts;
        sqs [b * BPB + blk] = tq;
    }
}

// ---------------------------------------------------------------------------
// Stage 2: fold 127 partials per batch (one wave per batch, fixed-order
// per-lane accumulation + fixed shuffle tree => deterministic).
// ---------------------------------------------------------------------------
__global__ void __launch_bounds__(32)
ipn_stage2(const float* __restrict__ sums,
           const float* __restrict__ sqs,
           float* __restrict__ stats)  // [B][4] = {mean, 1/(std+eps), std, -}
{
    const int b = blockIdx.x;
    const int t = threadIdx.x;  // 0..31

    float s = 0.0f, q = 0.0f;
    for (int k = t; k < BPB; k += 32) {
        s += sums[b * BPB + k];
        q += sqs [b * BPB + k];
    }
    for (int m = 16; m >= 1; m >>= 1) {
        s += __shfl_xor(s, m, 32);
        q += __shfl_xor(q, m, 32);
    }
    if (t == 0) {
        const float inv  = 1.0f / (float)ELEMS;
        const float mean = s * inv;
        const float var  = fmaxf(q * inv - mean * mean, 0.0f);
        const float sd   = sqrtf(var);
        const float rs   = 1.0f / (sd + EPS);
        stats[b * 4 + 0] = mean;
        stats[b * 4 + 1] = rs;
        stats[b * 4 + 2] = sd;
    }
}

// ---------------------------------------------------------------------------
// Stage 3: normalize + append {mean,std} + scale/bias.
// One block per row; thread t = output channel t. Non-temporal stores keep
// the 134 MB output stream from evicting x (133 MB, resident in 192 MB L2).
// ---------------------------------------------------------------------------
__global__ void __launch_bounds__(256)
ipn_stage3(const float* __restrict__ x,
           const float* __restrict__ w,
           const float* __restrict__ bias,
           const float* __restrict__ stats,
           float* __restrict__ out)
{
    const int row = blockIdx.x;       // 0 .. B*N-1
    const int b   = row / N;
    const int t   = threadIdx.x;      // 0 .. 255

    const float mean = stats[b * 4 + 0];
    const float rs   = stats[b * 4 + 1];
    const float sd   = stats[b * 4 + 2];

    float val;
    if (t < D)          val = (x[(size_t)row * D + t] - mean) * rs;
    else if (t == D)    val = mean;
    else                val = sd;

    const float o = val * w[t] + bias[t];
    __builtin_nontemporal_store(o, &out[(size_t)row * (D + 2) + t]);
}

// ---------------------------------------------------------------------------
// Host launcher
// ---------------------------------------------------------------------------
extern "C" void kernel_launch(void* const* d_in, const int* in_sizes, int n_in,
                              void* d_out, int out_size, void* d_ws, size_t ws_size,
                              hipStream_t stream) {
    (void)in_sizes; (void)n_in; (void)out_size; (void)ws_size;

    const float* x    = (const float*)d_in[0];
    const float* w    = (const float*)d_in[1];
    const float* bias = (const float*)d_in[2];
    float*       out  = (float*)d_out;

    // Workspace layout (floats): sums[B*BPB] | sqs[B*BPB] | stats[B*4]
    float* sums  = (float*)d_ws;
    float* sqs   = sums + (size_t)B * BPB;
    float* stats = sqs  + (size_t)B * BPB;

    ipn_stage1<<<B * BPB, THREADS, 0, stream>>>(x, sums, sqs);
    ipn_stage2<<<B, 32, 0, stream>>>(sums, sqs, stats);
    ipn_stage3<<<B * N, 256, 0, stream>>>(x, w, bias, stats, out);
}